// AttentionSplit_25134148616640
// MI455X (gfx1250) — compile-verified
//
#include <hip/hip_runtime.h>
#include <hip/hip_bf16.h>

#define BSZ 32
#define LSEQ 1024
#define NHEAD 8
#define DHEAD 64
#define HIDN 512
#define FINAL_N ((size_t)BSZ * LSEQ * HIDN)

typedef __attribute__((ext_vector_type(2))) float v2f;
typedef __attribute__((ext_vector_type(8))) float v8f;

__device__ __forceinline__ float geluf(float x) {
    return 0.5f * x * (1.0f + erff(x * 0.7071067811865475f));
}
__device__ __forceinline__ float seluf(float x) {
    const float alpha = 1.6732632423543772f;
    const float scale = 1.0507009873554805f;
    return scale * (x > 0.0f ? x : alpha * (expf(x) - 1.0f));
}
__device__ __forceinline__ float wave_sum(float v) {
#pragma unroll
    for (int o = 16; o > 0; o >>= 1) v += __shfl_xor(v, o, 32);
    return v;
}

// ---------------------------------------------------------------------------
// K1: block-diagonal 8x8 matmul + LN(64) + gelu  -> buf  (one block per (b,l))
// ---------------------------------------------------------------------------
__global__ void k1_headmm_ln(const float* __restrict__ x, const float* __restrict__ hw,
                             const float* __restrict__ fng, const float* __restrict__ fnb,
                             float* __restrict__ buf) {
    __shared__ float sx[512];
    __shared__ float shw[64];
    const int bl  = blockIdx.x;           // b*LSEQ + l
    const int tid = threadIdx.x;
    sx[tid]       = x[(size_t)bl * 512 + tid];
    sx[tid + 256] = x[(size_t)bl * 512 + tid + 256];
    if (tid < 64) shw[tid] = hw[tid];
    __syncthreads();

    const int h = tid >> 5, lane = tid & 31;
    float po[2];
#pragma unroll
    for (int e = 0; e < 2; e++) {
        const int j   = lane + e * 32;
        const int col = j & 7;
        const float* in8 = &sx[h * 64 + (j & 56)];
        float acc = 0.0f;
#pragma unroll
        for (int c = 0; c < 8; c++) acc = fmaf(in8[c], shw[c * 8 + col], acc);
        po[e] = acc;
    }
    const float mean = wave_sum(po[0] + po[1]) * (1.0f / 64.0f);
    const float d0 = po[0] - mean, d1 = po[1] - mean;
    const float rs = rsqrtf(wave_sum(d0 * d0 + d1 * d1) * (1.0f / 64.0f) + 1e-5f);
#pragma unroll
    for (int e = 0; e < 2; e++) {
        const int j = lane + e * 32;
        const float vv = (po[e] - mean) * rs * fng[j] + fnb[j];
        buf[(size_t)bl * 512 + h * 64 + j] = geluf(vv);
    }
}

// ---------------------------------------------------------------------------
// K2: elementwise time recurrence over l (16384 independent chains).
//     comb_l = po_l + gelu(t_{l-1}*s_h);  t_l = selu(comb_l)
//     buf <- comb (in place), ctxw <- t  (second half of d_out)
// ---------------------------------------------------------------------------
__global__ void k2_recur(const float* __restrict__ cells, float* __restrict__ buf,
                         float* __restrict__ ctxw) {
    const int g = blockIdx.x * blockDim.x + threadIdx.x;  // 0..16383
    const int b = g >> 9;
    const int r = g & 511;        // h*64 + j
    const int h = r >> 6;
    float s = 0.0f;
#pragma unroll 8
    for (int k = 0; k < 64; k++) s += cells[h * 64 + k];

    const size_t base = (size_t)b * LSEQ * 512 + r;
    float t  = 0.0f;
    float po = buf[base];                       // prefetch l=0
    for (int l = 0; l < LSEQ; l++) {
        const float po_n = (l + 1 < LSEQ) ? buf[base + (size_t)(l + 1) * 512] : 0.0f;
        const float comb = po + geluf(t * s);
        const float nt   = seluf(comb);
        buf[base + (size_t)l * 512]  = comb;
        ctxw[base + (size_t)l * 512] = nt;
        t  = nt;
        po = po_n;
    }
}

// ---------------------------------------------------------------------------
// K3: LN(64) + gelu over comb, in place (one wave per (b,l,h))
// ---------------------------------------------------------------------------
__global__ void k3_ln64(float* __restrict__ buf, const float* __restrict__ cng,
                        const float* __restrict__ cnb) {
    const int tid = threadIdx.x;
    const int w = tid >> 5, lane = tid & 31;
    const size_t base = ((size_t)blockIdx.x * 8 + w) * 64;
    const float v0 = buf[base + lane];
    const float v1 = buf[base + lane + 32];
    const float mean = wave_sum(v0 + v1) * (1.0f / 64.0f);
    const float d0 = v0 - mean, d1 = v1 - mean;
    const float rs = rsqrtf(wave_sum(d0 * d0 + d1 * d1) * (1.0f / 64.0f) + 1e-5f);
    buf[base + lane]      = geluf(d0 * rs * cng[lane] + cnb[lane]);
    buf[base + lane + 32] = geluf(d1 * rs * cng[lane + 32] + cnb[lane + 32]);
}

// ---------------------------------------------------------------------------
// K4: 7x7 conv (1->8 ch) via WMMA f32 16x16x4 implicit GEMM + bias + gelu,
//     then channel recurrence (7 selu steps), LN(512), gelu -> d_out.
//     One block per (b, y) output row; 8 waves x 4 tiles of 16 pixels.
//     Input halo tile is staged with gfx1250 async global->LDS b128 loads.
// ---------------------------------------------------------------------------
__global__ void k4_conv_wmma(const float* __restrict__ seq, const float* __restrict__ cw,
                             const float* __restrict__ cb, const float* __restrict__ lng,
                             const float* __restrict__ lnb, float* __restrict__ outp) {
    __shared__ float s_in[7][524];   // global x stored at col = x+4 (16B-aligned rows)
    __shared__ float s_w[8][52];     // zero-padded taps (49 -> 52)
    __shared__ float s_co[8][512];   // gelu(conv + bias)
    __shared__ float s_red[16];

    const int tid = threadIdx.x;
    const int b = blockIdx.x >> 10;
    const int y = blockIdx.x & 1023;

    // zero-fill: halo columns of every row, plus out-of-range rows entirely
    for (int i = tid; i < 7 * 524; i += 256) {
        const int row = i / 524, col = i % 524;
        const int yy = y + row - 3;
        if (yy < 0 || yy >= LSEQ || col < 4 || col >= 516) s_in[row][col] = 0.0f;
    }
    // async global->LDS: 128 x b128 per valid row (interior cols 4..515)
    for (int i = tid; i < 7 * 128; i += 256) {
        const int row = i >> 7, q = i & 127;     // q = 16B chunk within row
        const int yy = y + row - 3;
        if (yy >= 0 && yy < LSEQ) {
            const unsigned lds_off =
                (unsigned)(uintptr_t)(&s_in[row][4 + 4 * q]);
            const unsigned long long gaddr =
                (unsigned long long)(uintptr_t)(seq + ((size_t)b * LSEQ + yy) * 512 + 4 * q);
            asm volatile("global_load_async_to_lds_b128 %0, %1, off"
                         :: "v"(lds_off), "v"(gaddr)
                         : "memory");
        }
    }
    for (int i = tid; i < 8 * 52; i += 256) {
        const int o = i / 52, t = i % 52;
        s_w[o][t] = (t < 49) ? cw[o * 49 + t] : 0.0f;
    }
    asm volatile("s_wait_asynccnt 0x0" ::: "memory");
    __syncthreads();

    const int w = tid >> 5, lane = tid & 31;
    const int m = lane & 15;               // pixel (A row) / channel (B col)
    const int khalf = (lane >> 4) * 2;     // lanes 0-15: K 0/1, lanes 16-31: K 2/3
#pragma unroll
    for (int tile = 0; tile < 4; tile++) {
        const int x0 = (w * 4 + tile) * 16;
        v8f acc = {0.0f, 0.0f, 0.0f, 0.0f, 0.0f, 0.0f, 0.0f, 0.0f};
#pragma unroll
        for (int chunk = 0; chunk < 13; chunk++) {
            v2f a, bb;
#pragma unroll
            for (int p = 0; p < 2; p++) {
                const int t  = chunk * 4 + khalf + p;
                const int tc = t < 49 ? t : 48;     // clamp reads; B pad is 0
                const int dy = tc / 7, dx = tc % 7;
                a[p]  = s_in[dy][x0 + m + dx + 1];  // gx = x0+m+dx-3, stored at gx+4
                bb[p] = (m < 8) ? s_w[m][t] : 0.0f; // channels 8..15 are zero pad
            }
            acc = __builtin_amdgcn_wmma_f32_16x16x4_f32(
                false, a, false, bb, (short)0, acc, false, false);
        }
        if (m < 8) {                       // lane holds channel n = m
            const float bias = cb[m];
#pragma unroll
            for (int vv = 0; vv < 8; vv++) {
                const int px = x0 + vv + (lane >> 4) * 8;  // D vgpr v -> M = v / v+8
                s_co[m][px] = geluf(acc[vv] + bias);
            }
        }
    }
    __syncthreads();

    // channel recurrence: c_o = selu(co_o - c_{o-1}), c_0 = co_0 ; last = c_7
    float last[2];
#pragma unroll
    for (int e = 0; e < 2; e++) {
        const int xx = tid + e * 256;
        float c = s_co[0][xx];
#pragma unroll
        for (int o = 1; o < 8; o++) c = seluf(s_co[o][xx] - c);
        last[e] = c;
    }
    const float wsum = wave_sum(last[0] + last[1]);
    const float wsq  = wave_sum(last[0] * last[0] + last[1] * last[1]);
    if (lane == 0) { s_red[w] = wsum; s_red[8 + w] = wsq; }
    __syncthreads();
    float sum = 0.0f, sq = 0.0f;
#pragma unroll
    for (int i = 0; i < 8; i++) { sum += s_red[i]; sq += s_red[8 + i]; }
    const float mean = sum * (1.0f / 512.0f);
    const float var  = sq * (1.0f / 512.0f) - mean * mean;
    const float rs   = rsqrtf(var + 1e-5f);
#pragma unroll
    for (int e = 0; e < 2; e++) {
        const int xx = tid + e * 256;
        const float vv = (last[e] - mean) * rs * lng[xx] + lnb[xx];
        outp[((size_t)b * LSEQ + y) * 512 + xx] = geluf(vv);
    }
}

// ---------------------------------------------------------------------------
extern "C" void kernel_launch(void* const* d_in, const int* in_sizes, int n_in,
                              void* d_out, int out_size, void* d_ws, size_t ws_size,
                              hipStream_t stream) {
    const float* inputs = (const float*)d_in[0];
    const float* cells  = (const float*)d_in[1];
    const float* hw     = (const float*)d_in[2];
    const float* cw     = (const float*)d_in[3];
    const float* cb     = (const float*)d_in[4];
    const float* fng    = (const float*)d_in[5];
    const float* fnb    = (const float*)d_in[6];
    const float* cng    = (const float*)d_in[7];
    const float* cnb    = (const float*)d_in[8];
    const float* lng    = (const float*)d_in[9];
    const float* lnb    = (const float*)d_in[10];

    float* out  = (float*)d_out;
    float* ctxw = out + FINAL_N;           // second return value, same flat layout
    float* buf  = (float*)d_ws;            // 64 MB, reused in place across stages

    k1_headmm_ln<<<BSZ * LSEQ, 256, 0, stream>>>(inputs, hw, fng, fnb, buf);
    k2_recur    <<<64,         256, 0, stream>>>(cells, buf, ctxw);
    k3_ln64     <<<BSZ * LSEQ, 256, 0, stream>>>(buf, cng, cnb);
    k4_conv_wmma<<<BSZ * LSEQ, 256, 0, stream>>>(buf, cw, cb, lng, lnb, out);
}